// NN_Model_47734266528217
// MI455X (gfx1250) — compile-verified
//
#include <hip/hip_runtime.h>
#include <math.h>

typedef __attribute__((ext_vector_type(2))) float v2f;
typedef __attribute__((ext_vector_type(8))) float v8f;

#define LEN_M  131328          // 513 * 256
#define D_H    256
#define N_MC   64
#define DDIM   513             // DATA_DIM_P1
#define NROWS  4096

// ---------------------------------------------------------------------------
// Kernel 1: prep.  m_w0 / v_w0 straight to the output tuple, s = sqrt(v) to ws.
// ---------------------------------------------------------------------------
__global__ void __launch_bounds__(256)
prep_kernel(const float* __restrict__ params,
            float* __restrict__ out_m,
            float* __restrict__ out_v,
            float* __restrict__ s_buf) {
    int i = blockIdx.x * blockDim.x + threadIdx.x;
    if (i < LEN_M) {
        float m  = params[i];
        float vr = params[LEN_M + i];
        float v  = (vr > 0.0f ? vr : 0.0f) + 1e-6f;
        out_m[i] = m;
        out_v[i] = v;
        s_buf[i] = sqrtf(v);
    }
}

// ---------------------------------------------------------------------------
// Kernel 2: dual f32 WMMA GEMM.
//   A[n,h] = sum_d x[n,d] * m[d,h]
//   B[n,h] = sum_d x[n,d] * s[d,h]
// One wave (32 lanes) owns one 16x16 tile of both outputs.
// V_WMMA_F32_16X16X4_F32 fragment layout (ISA 7.12.2, wave32):
//   A (16x4):  lane%16 = M row; lane/16 selects K pair; vgpr0=K even, vgpr1=K odd
//   B (4x16):  lane%16 = N col; lane/16 selects K pair; vgpr0=K even, vgpr1=K odd
//   C/D (16x16): vgpr r -> M = r + 8*(lane/16); lane%16 -> N
// K = 513 = 128*4 + 1  -> 128 WMMA steps + scalar rank-1 remainder (d = 512).
// ---------------------------------------------------------------------------
__global__ void __launch_bounds__(256)
gemm_kernel(const float* __restrict__ x,     // [4096, 513]
            const float* __restrict__ wm,    // m_w0 viewed as [513, 256]
            const float* __restrict__ ws,    // s    viewed as [513, 256]
            float* __restrict__ Abuf,        // [4096, 256]
            float* __restrict__ Bbuf) {      // [4096, 256]
    const int wave  = (blockIdx.x * blockDim.x + threadIdx.x) >> 5;
    const int lane  = threadIdx.x & 31;
    const int tileH = wave & 15;             // 256/16 = 16 column tiles
    const int tileM = wave >> 4;             // 4096/16 = 256 row tiles
    const int mrow  = tileM * 16 + (lane & 15);   // A-fragment row for this lane
    const int col   = tileH * 16 + (lane & 15);   // B/C/D-fragment column
    const int hi    = lane >> 4;                  // K-pair select (0 or 1)

    const float* __restrict__ xrow = x + (long)mrow * DDIM;

    v8f accM = {};
    v8f accS = {};

    for (int k = 0; k < 512; k += 4) {
        const int ka = k + 2 * hi;           // this lane's K pair: ka, ka+1
        v2f a, bm, bs;
        a.x  = xrow[ka];
        a.y  = xrow[ka + 1];
        bm.x = wm[(long)ka * D_H + col];
        bm.y = wm[(long)(ka + 1) * D_H + col];
        bs.x = ws[(long)ka * D_H + col];
        bs.y = ws[(long)(ka + 1) * D_H + col];
        // D = A*B + C  (f32 in, f32 accumulate)
        accM = __builtin_amdgcn_wmma_f32_16x16x4_f32(
                   false, a, false, bm, (short)0, accM, false, false);
        accS = __builtin_amdgcn_wmma_f32_16x16x4_f32(
                   false, a, false, bs, (short)0, accS, false, false);
    }

    // Remainder column d = 512: rank-1 update applied directly in C/D layout.
    const float wmc = wm[(long)512 * D_H + col];
    const float wsc = ws[(long)512 * D_H + col];
#pragma unroll
    for (int r = 0; r < 8; ++r) {
        const int rowc = tileM * 16 + r + 8 * hi;
        const float xv = x[(long)rowc * DDIM + 512];
        accM[r] += xv * wmc;
        accS[r] += xv * wsc;
    }

    // Store both tiles.
#pragma unroll
    for (int r = 0; r < 8; ++r) {
        const int rowc = tileM * 16 + r + 8 * hi;
        Abuf[(long)rowc * D_H + col] = accM[r];
        Bbuf[(long)rowc * D_H + col] = accS[r];
    }
}

// ---------------------------------------------------------------------------
// Kernel 3: epilogue.
//   pred[n,mc] = W1[0] + sum_h relu(A[n,h] + eps[mc]*B[n,h]) * W1[h+1]
// One 64-thread block (= 2 waves) per row n; thread = MC sample.
// A/B row elements are wave-uniform loads (broadcast), eps stays in a register.
// ---------------------------------------------------------------------------
__global__ void __launch_bounds__(64)
pred_kernel(const float* __restrict__ Abuf,
            const float* __restrict__ Bbuf,
            const float* __restrict__ W1,    // [257]
            const float* __restrict__ eps,   // [64]
            float* __restrict__ pred) {      // [4096, 64]
    const int n  = blockIdx.x;
    const int mc = threadIdx.x;              // 0..63
    const float e = eps[mc];
    const float* __restrict__ Ar = Abuf + (long)n * D_H;
    const float* __restrict__ Br = Bbuf + (long)n * D_H;

    float acc = 0.0f;
#pragma unroll 8
    for (int h = 0; h < D_H; ++h) {
        float t = fmaf(e, Br[h], Ar[h]);
        t = t > 0.0f ? t : 0.0f;
        acc = fmaf(t, W1[h + 1], acc);
    }
    pred[(long)n * N_MC + mc] = acc + W1[0];
}

// ---------------------------------------------------------------------------
// Host-side launcher.
// Inputs (setup_inputs order): params[2*LEN_M], W1[257], x[4096*513], eps[64]
// Output tuple, flattened: pred[4096*64] | m_w0[LEN_M] | v_w0[LEN_M]
// Workspace: s[LEN_M] | Abuf[4096*256] | Bbuf[4096*256]  (~8.9 MB)
// ---------------------------------------------------------------------------
extern "C" void kernel_launch(void* const* d_in, const int* in_sizes, int n_in,
                              void* d_out, int out_size, void* d_ws, size_t ws_size,
                              hipStream_t stream) {
    (void)in_sizes; (void)n_in; (void)out_size; (void)ws_size;

    const float* params = (const float*)d_in[0];
    const float* W1     = (const float*)d_in[1];
    const float* x      = (const float*)d_in[2];
    const float* eps    = (const float*)d_in[3];

    float* pred  = (float*)d_out;
    float* out_m = pred + (long)NROWS * N_MC;
    float* out_v = out_m + LEN_M;

    float* s_buf = (float*)d_ws;
    float* Abuf  = s_buf + LEN_M;
    float* Bbuf  = Abuf + (long)NROWS * D_H;

    // 1) prep: m/v outputs + s = sqrt(v)
    prep_kernel<<<(LEN_M + 255) / 256, 256, 0, stream>>>(params, out_m, out_v, s_buf);

    // 2) dual WMMA GEMM: 4096 tiles -> 4096 waves -> 512 blocks of 8 waves
    gemm_kernel<<<512, 256, 0, stream>>>(x, params /* m_w0 */, s_buf, Abuf, Bbuf);

    // 3) epilogue reduction
    pred_kernel<<<NROWS, 64, 0, stream>>>(Abuf, Bbuf, W1, eps, pred);
}